// SimGCL_61632780698012
// MI455X (gfx1250) — compile-verified
//
#include <hip/hip_runtime.h>
#include <hip/hip_bf16.h>
#include <stdint.h>

// ---------------------------------------------------------------------------
// LightGCN propagation on MI455X (gfx1250).
// Memory/atomic-bound SpMM scatter: A is random COO (no dense tiles) -> WMMA
// is inapplicable; the chip features that pay here are the CDNA5 async
// global->LDS path (ASYNCcnt) for staging COO triples, wave32 mapping with
// 2 nonzeros/wave (16 lanes x float4 = 256B row), and fp32 atomics resolved
// in the 192MB L2 (entire ~165MB working set is L2-resident).
// ---------------------------------------------------------------------------

#define EMBED_DIM 64
#define TILE      1024   // nonzeros staged per block
#define BLK       256    // 8 waves of 32

// ---- init: x_cur = concat(user_w, item_w); acc = x_cur; x_next = 0 --------
__global__ void k_init(const float4* __restrict__ uw, const float4* __restrict__ iw,
                       float4* __restrict__ xcur, float4* __restrict__ xnext,
                       float4* __restrict__ acc, int n4, int nu4) {
    int i = blockIdx.x * blockDim.x + threadIdx.x;
    if (i >= n4) return;
    float4 v = (i < nu4) ? uw[i] : iw[i - nu4];
    xcur[i] = v;
    acc[i]  = v;
    float4 z; z.x = 0.f; z.y = 0.f; z.z = 0.f; z.w = 0.f;
    xnext[i] = z;
}

// ---- fused: acc += x_new ; zero the buffer that becomes the next target ---
__global__ void k_accum(float4* __restrict__ acc, const float4* __restrict__ xnew,
                        float4* __restrict__ xzero, int n4, int do_scale) {
    int i = blockIdx.x * blockDim.x + threadIdx.x;
    if (i >= n4) return;
    float4 a = acc[i];
    float4 b = xnew[i];
    a.x += b.x; a.y += b.y; a.z += b.z; a.w += b.w;
    if (do_scale) { a.x *= 0.25f; a.y *= 0.25f; a.z *= 0.25f; a.w *= 0.25f; }
    acc[i] = a;
    float4 z; z.x = 0.f; z.y = 0.f; z.z = 0.f; z.w = 0.f;
    xzero[i] = z;
}

// ---- SpMM: xdst[r,:] += v * xsrc[c,:] for each COO nonzero ----------------
// COO triples for a TILE are bulk-staged into LDS with async b128 loads
// (gfx1250 ASYNCcnt path). Each wave then processes 2 nonzeros per iteration:
// 16 lanes per nonzero, float4 per lane -> one global_load_b128 gather per
// lane covering the 256B row, then 4 no-return f32 atomics per lane.
__global__ void k_spmm(const float* __restrict__ vals, const int* __restrict__ rows,
                       const int* __restrict__ cols, const float* __restrict__ xsrc,
                       float* __restrict__ xdst, int nnz) {
    __shared__ uint32_t smem[3 * TILE];          // 4KB vals | 4KB rows | 4KB cols
    const int t     = threadIdx.x;               // 0..255
    const int tile0 = blockIdx.x * TILE;

    // ---- async stage: 256 threads x 3 arrays x 16B = 12KB ----
    {
        uint32_t ldsbase = (uint32_t)(uintptr_t)(&smem[0]);
        // clamp global offset so tail blocks never read OOB (slots guarded later)
        long maxoff = ((long)nnz * 4 - 16) & ~15L;
        long want   = (long)tile0 * 4 + (long)t * 16;
        uint32_t goff = (uint32_t)(want > maxoff ? maxoff : want);
        uint32_t l0 = ldsbase + (uint32_t)t * 16u;        // vals slot
        uint32_t l1 = l0 + (uint32_t)(TILE * 4);          // rows slot
        uint32_t l2 = l0 + (uint32_t)(2 * TILE * 4);      // cols slot
        asm volatile("global_load_async_to_lds_b128 %0, %1, %2"
                     :: "v"(l0), "v"(goff), "s"(vals) : "memory");
        asm volatile("global_load_async_to_lds_b128 %0, %1, %2"
                     :: "v"(l1), "v"(goff), "s"(rows) : "memory");
        asm volatile("global_load_async_to_lds_b128 %0, %1, %2"
                     :: "v"(l2), "v"(goff), "s"(cols) : "memory");
        asm volatile("s_wait_asynccnt 0" ::: "memory");
    }
    __syncthreads();

    const int wave = t >> 5;        // wave32 on CDNA5
    const int lane = t & 31;
    const int half = lane >> 4;     // which of the 2 nonzeros this lane serves
    const int l16  = lane & 15;     // 0..15 -> float4 slot within the 64-wide row
    const float* smem_v = (const float*)smem;
    const int*   smem_r = (const int*)(smem + TILE);
    const int*   smem_c = (const int*)(smem + 2 * TILE);

    for (int j2 = wave * 2; j2 < TILE; j2 += 16) {   // 8 waves x 2 nnz/iter
        if (tile0 + j2 >= nnz) break;                // wave-uniform exit
        int j = j2 + half;
        int g = tile0 + j;
        if (g < nnz) {                               // per-lane tail guard
            float v = smem_v[j];                     // LDS broadcast (per half-wave)
            int   r = smem_r[j];
            int   c = smem_c[j];
            const float4* srow = (const float4*)(xsrc + (size_t)c * EMBED_DIM);
            float4 xv = srow[l16];                   // global_load_b128, coalesced
            float* drow = xdst + (size_t)r * EMBED_DIM + (size_t)(l16 * 4);
            atomicAdd(drow + 0, v * xv.x);           // global_atomic_add_f32
            atomicAdd(drow + 1, v * xv.y);
            atomicAdd(drow + 2, v * xv.z);
            atomicAdd(drow + 3, v * xv.w);
        }
    }
}

// ---------------------------------------------------------------------------
extern "C" void kernel_launch(void* const* d_in, const int* in_sizes, int n_in,
                              void* d_out, int out_size, void* d_ws, size_t ws_size,
                              hipStream_t stream) {
    const float* user_w = (const float*)d_in[0];
    const float* item_w = (const float*)d_in[1];
    const float* vals   = (const float*)d_in[2];
    const int*   rows   = (const int*)d_in[3];
    const int*   cols   = (const int*)d_in[4];

    const int nu  = in_sizes[0] / EMBED_DIM;
    const int ni  = in_sizes[1] / EMBED_DIM;
    const int n   = nu + ni;
    const int nnz = in_sizes[2];

    const size_t NF = (size_t)n * EMBED_DIM;     // floats per embedding buffer
    float* acc  = (float*)d_out;                 // accumulator == output layout
    float* bufA = (float*)d_ws;                  // ping
    float* bufB = bufA + NF;                     // pong

    const int n4    = (int)(NF / 4);
    const int nu4   = in_sizes[0] / 4;
    const int gridE = (n4 + BLK - 1) / BLK;
    const int tiles = (nnz + TILE - 1) / TILE;

    // acc = x0 ; bufA = x0 ; bufB = 0
    k_init<<<gridE, BLK, 0, stream>>>((const float4*)user_w, (const float4*)item_w,
                                      (float4*)bufA, (float4*)bufB, (float4*)acc,
                                      n4, nu4);
    // layer 1: x1 = A x0
    k_spmm<<<tiles, BLK, 0, stream>>>(vals, rows, cols, bufA, bufB, nnz);
    k_accum<<<gridE, BLK, 0, stream>>>((float4*)acc, (const float4*)bufB,
                                       (float4*)bufA, n4, 0);
    // layer 2: x2 = A x1
    k_spmm<<<tiles, BLK, 0, stream>>>(vals, rows, cols, bufB, bufA, nnz);
    k_accum<<<gridE, BLK, 0, stream>>>((float4*)acc, (const float4*)bufA,
                                       (float4*)bufB, n4, 0);
    // layer 3: x3 = A x2 ; acc = (acc + x3) / 4
    k_spmm<<<tiles, BLK, 0, stream>>>(vals, rows, cols, bufA, bufB, nnz);
    k_accum<<<gridE, BLK, 0, stream>>>((float4*)acc, (const float4*)bufB,
                                       (float4*)bufA, n4, 1);
}